// PatchNorm_58420145160519
// MI455X (gfx1250) — compile-verified
//
#include <hip/hip_runtime.h>
#include <cstdint>
#include <cstddef>

// ---------------------------------------------------------------------------
// PatchNorm (batched Welford scatter + normalize) for MI455X / gfx1250.
// Memory-bound: ~128MiB in + ~128MiB out => ~11us floor @ 23.3 TB/s; patches
// (128MiB) stays resident in the 192MB L2 across passes.
// Counting-sort tokens by (c,h,w) bin; one workgroup per bin reduces
// sum(d) and sum(d^2) in a single token pass with V_WMMA_F32_16X16X4_F32
// (A = ones, exact f32: D[m,n] = C[m,n] + sum_k B[k,n]); token ids staged in
// LDS to break the gather dependency chain. No atomics in heavy passes.
// Normalize pass: global_load_async_to_lds_b128 + s_wait_asynccnt,
// global_prefetch_b8, nontemporal b128 stores.
// ---------------------------------------------------------------------------

#define CC   3
#define HH   32
#define WW   32
#define ZZ   256
#define BB   128
#define SS   1024
#define NTOK (BB * SS)        // 131072 tokens
#define NBIN (CC * HH * WW)   // 3072 bins
#define NME  (NBIN * ZZ)      // 786432 stat elements
#define NEL  (NTOK * ZZ)      // 33554432 output elements
#define NQ   (NEL / 4)        // 8388608 float4 quads
#define EPSV 1e-6f
#define LIDS 4096             // staged token ids per bin (16KB LDS)

typedef float f2 __attribute__((ext_vector_type(2)));
typedef float f4 __attribute__((ext_vector_type(4)));
typedef float f8 __attribute__((ext_vector_type(8)));

// workspace layout in 4-byte units
#define WS_COUNTS 0                    // 3072 u32
#define WS_BSTART 3072                 // 3072 u32 (exclusive prefix)
#define WS_CURSOR 6144                 // 3072 u32 (scatter cursors)
#define WS_NNEW   9216                 // 3072 f32 (n + count)
#define WS_FLAT   12288                // 131072 u32 per-token bin
#define WS_VALID  (WS_FLAT + NTOK)     // 131072 f32
#define WS_IDS    (WS_VALID + NTOK)    // 131072 u32 bin-sorted token ids
#define WS_MEAN   (WS_IDS + NTOK)      // 786432 f32 mean_new
#define WS_STD    (WS_MEAN + NME)      // 786432 f32 std
// total: 1,978,368 dwords = 7.55 MiB

// ---- K1: zero counts; precompute per-token flat bin + valid ---------------
__global__ void __launch_bounds__(256) k_init(
    const int* __restrict__ pc, const int* __restrict__ ph,
    const int* __restrict__ pw, const unsigned char* __restrict__ mask,
    unsigned* __restrict__ wsu, float* __restrict__ wsf) {
  int i = blockIdx.x * 256 + threadIdx.x;          // grid covers NTOK exactly
  if (i < NBIN) wsu[WS_COUNTS + i] = 0u;
  unsigned fl = (unsigned)(pc[i] * (HH * WW) + ph[i] * WW + pw[i]);
  wsu[WS_FLAT + i] = fl;
  wsf[WS_VALID + i] = mask[i] ? 0.0f : 1.0f;
}

// ---- K2: histogram of valid tokens per bin --------------------------------
__global__ void __launch_bounds__(256) k_count(unsigned* __restrict__ wsu,
                                               const float* __restrict__ wsf) {
  int i = blockIdx.x * 256 + threadIdx.x;          // grid covers NTOK exactly
  if (wsf[WS_VALID + i] != 0.0f)
    atomicAdd(&wsu[WS_COUNTS + wsu[WS_FLAT + i]], 1u);
}

// ---- K3: single-block exclusive scan of counts; init cursors; n_new -------
__global__ void __launch_bounds__(256) k_scan(const float* __restrict__ n,
                                              unsigned* __restrict__ wsu,
                                              float* __restrict__ wsf) {
  __shared__ unsigned sh[256];
  int t = threadIdx.x;                             // 256 threads x 12 bins
  unsigned loc[12], s = 0;
  for (int j = 0; j < 12; ++j) { loc[j] = wsu[WS_COUNTS + t * 12 + j]; s += loc[j]; }
  sh[t] = s;
  __syncthreads();
  for (int off = 1; off < 256; off <<= 1) {        // Hillis-Steele inclusive
    unsigned v = (t >= off) ? sh[t - off] : 0u;
    __syncthreads();
    sh[t] += v;
    __syncthreads();
  }
  unsigned ex = sh[t] - s;                         // exclusive prefix of chunk
  for (int j = 0; j < 12; ++j) {
    int bin = t * 12 + j;
    wsu[WS_BSTART + bin] = ex;
    wsu[WS_CURSOR + bin] = ex;
    wsf[WS_NNEW + bin] = n[bin] + (float)loc[j];
    ex += loc[j];
  }
}

// ---- K4: bucket valid token ids by bin ------------------------------------
__global__ void __launch_bounds__(256) k_scatter(unsigned* __restrict__ wsu,
                                                 const float* __restrict__ wsf) {
  int i = blockIdx.x * 256 + threadIdx.x;          // grid covers NTOK exactly
  if (wsf[WS_VALID + i] != 0.0f) {
    unsigned pos = atomicAdd(&wsu[WS_CURSOR + wsu[WS_FLAT + i]], 1u);
    wsu[WS_IDS + pos] = (unsigned)i;
  }
}

// ---- K5 body: single-pass sum(d), sum(d^2) per column via WMMA ------------
// A = ones(16x4 f32) => D[m,n] = C[m,n] + sum_k B[k,n]. With A all-ones the
// column sum is layout-robust: it is the sum of the 4 B values held in
// {vgpr0,vgpr1} x {lane, lane^16}, which is exactly how we pack tokens
// t..t+3. All D rows are equal; acc[0] in each lane = column (lane%16) sum.
template <bool USE_LDS>
__device__ __forceinline__ void stats_body(
    const float* __restrict__ patches, const float* __restrict__ mean_old,
    const float* __restrict__ m2_old, const unsigned* __restrict__ gids,
    const unsigned* __restrict__ lids, unsigned start, unsigned cnt,
    float nnr, float nn, int bin, int wv, int col, unsigned half,
    float* __restrict__ wsf) {
  const f2 ones = {1.0f, 1.0f};
  const size_t rowb = (size_t)bin * ZZ;
  const int zA = (wv << 4) + col;                  // slice A: z in [0,128)
  const int zB = zA + 128;                         // slice B: z in [128,256)
  const float moA = mean_old[rowb + zA];
  const float moB = mean_old[rowb + zB];
  f8 sA = {}, qA = {}, sB = {}, qB = {};
#pragma unroll 2
  for (unsigned t = 0; t < cnt; t += 4) {
    unsigned i0 = t + half, i1 = t + 2 + half;
    float a0 = 0.0f, a1 = 0.0f, b0 = 0.0f, b1 = 0.0f;
    if (i0 < cnt) {
      unsigned tk = USE_LDS ? lids[i0] : gids[start + i0];
      const float* row = patches + (size_t)tk * ZZ;
      a0 = row[zA] - moA;
      b0 = row[zB] - moB;
    }
    if (i1 < cnt) {
      unsigned tk = USE_LDS ? lids[i1] : gids[start + i1];
      const float* row = patches + (size_t)tk * ZZ;
      a1 = row[zA] - moA;
      b1 = row[zB] - moB;
    }
    f2 dA = {a0, a1}, dB = {b0, b1};
    f2 eA = {a0 * a0, a1 * a1}, eB = {b0 * b0, b1 * b1};
    sA = __builtin_amdgcn_wmma_f32_16x16x4_f32(false, ones, false, dA, (short)0, sA, false, false);
    qA = __builtin_amdgcn_wmma_f32_16x16x4_f32(false, ones, false, eA, (short)0, qA, false, false);
    sB = __builtin_amdgcn_wmma_f32_16x16x4_f32(false, ones, false, dB, (short)0, sB, false, false);
    qB = __builtin_amdgcn_wmma_f32_16x16x4_f32(false, ones, false, eB, (short)0, qB, false, false);
  }
  // c = sum(d)/n_new; mean_new = mo + c; m2 += sum(d^2) - c*sum(d)
  float cA = sA[0] / nn, cB = sB[0] / nn;
  float m2A = m2_old[rowb + zA] + (qA[0] - cA * sA[0]);
  float m2B = m2_old[rowb + zB] + (qB[0] - cB * sB[0]);
  float vA = (nnr < 2.0f) ? 1.0f : m2A / nn;
  float vB = (nnr < 2.0f) ? 1.0f : m2B / nn;
  if (half == 0) {                                 // lane halves hold duplicates
    wsf[WS_MEAN + rowb + zA] = moA + cA;
    wsf[WS_MEAN + rowb + zB] = moB + cB;
    wsf[WS_STD + rowb + zA] = sqrtf(vA);
    wsf[WS_STD + rowb + zB] = sqrtf(vB);
  }
}

// ---- K5: one block per bin; ids staged in LDS; 8 waves x 2 z-slices -------
__global__ void __launch_bounds__(256) k_stats(
    const float* __restrict__ patches, const float* __restrict__ mean_old,
    const float* __restrict__ m2_old, const unsigned* __restrict__ wsu,
    float* __restrict__ wsf) {
  __shared__ unsigned lids[LIDS];
  const int bin  = blockIdx.x;
  const int lane = threadIdx.x & 31;
  const int wv   = threadIdx.x >> 5;
  const int col  = lane & 15;
  const unsigned half = (unsigned)(lane >> 4);

  const unsigned start = wsu[WS_BSTART + bin];
  const unsigned cnt   = wsu[WS_COUNTS + bin];
  for (unsigned i = threadIdx.x; i < cnt && i < LIDS; i += 256)
    lids[i] = wsu[WS_IDS + start + i];             // coalesced one-time stage
  __syncthreads();

  const float nnr = wsf[WS_NNEW + bin];
  const float nn  = fmaxf(nnr, 1.0f);
  if (cnt <= LIDS)                                 // uniform per block
    stats_body<true>(patches, mean_old, m2_old, wsu + WS_IDS, lids,
                     start, cnt, nnr, nn, bin, wv, col, half, wsf);
  else
    stats_body<false>(patches, mean_old, m2_old, wsu + WS_IDS, lids,
                      start, cnt, nnr, nn, bin, wv, col, half, wsf);
}

// ---- K6: normalize. Async-stage patch tile through LDS, overlap gathers ---
__global__ void __launch_bounds__(256) k_norm(
    const float* __restrict__ patches, const unsigned* __restrict__ wsu,
    const float* __restrict__ wsf, float* __restrict__ out) {
  __shared__ float tile[1024];                     // 4 tokens * 256 f32
  int gid = blockIdx.x * 256 + threadIdx.x;        // grid covers NQ exactly
  int tok = gid >> 6;
  int q   = (gid & 63) << 2;

  // async global->LDS copy of this lane's b128 chunk (ASYNCcnt path)
  const float* src = patches + (size_t)tok * ZZ + q;
  unsigned lds_off = (unsigned)(size_t)&tile[threadIdx.x * 4];
  asm volatile("global_load_async_to_lds_b128 %0, %1, off"
               :: "v"(lds_off), "v"(src) : "memory");

  // overlap: per-token scalars and mean/std row gathers
  unsigned fl = wsu[WS_FLAT + tok];
  float vv = wsf[WS_VALID + tok];
  const float* mrow = wsf + WS_MEAN + (size_t)fl * ZZ + q;
  const float* srow = wsf + WS_STD  + (size_t)fl * ZZ + q;
  __builtin_prefetch(mrow, 0, 1);                  // global_prefetch_b8
  __builtin_prefetch(srow, 0, 1);
  f4 mn = *(const f4*)mrow;
  f4 sd = *(const f4*)srow;

  asm volatile("s_wait_asynccnt 0x0" ::: "memory");
  f4 p = *(const f4*)&tile[threadIdx.x * 4];

  f4 r;
  r.x = fminf(fmaxf((p.x - mn.x) / (sd.x + EPSV), -5.0f), 5.0f) * vv;
  r.y = fminf(fmaxf((p.y - mn.y) / (sd.y + EPSV), -5.0f), 5.0f) * vv;
  r.z = fminf(fmaxf((p.z - mn.z) / (sd.z + EPSV), -5.0f), 5.0f) * vv;
  r.w = fminf(fmaxf((p.w - mn.w) / (sd.w + EPSV), -5.0f), 5.0f) * vv;
  __builtin_nontemporal_store(r, (f4*)(out + (size_t)tok * ZZ + q));
}

extern "C" void kernel_launch(void* const* d_in, const int* in_sizes, int n_in,
                              void* d_out, int out_size, void* d_ws, size_t ws_size,
                              hipStream_t stream) {
  (void)in_sizes; (void)n_in; (void)out_size; (void)ws_size;
  const float*         patches = (const float*)d_in[0];
  const int*           pc      = (const int*)d_in[1];
  const int*           ph      = (const int*)d_in[2];
  const int*           pw      = (const int*)d_in[3];
  const unsigned char* mask    = (const unsigned char*)d_in[4];  // jnp.bool_
  const float*         n       = (const float*)d_in[5];
  const float*         mean    = (const float*)d_in[6];
  const float*         m2      = (const float*)d_in[7];
  float*    out = (float*)d_out;
  float*    wsf = (float*)d_ws;     // needs ~7.6 MiB
  unsigned* wsu = (unsigned*)d_ws;

  k_init   <<<NTOK / 256, 256, 0, stream>>>(pc, ph, pw, mask, wsu, wsf);
  k_count  <<<NTOK / 256, 256, 0, stream>>>(wsu, wsf);
  k_scan   <<<1,          256, 0, stream>>>(n, wsu, wsf);
  k_scatter<<<NTOK / 256, 256, 0, stream>>>(wsu, wsf);
  k_stats  <<<NBIN,       256, 0, stream>>>(patches, mean, m2, wsu, wsf);
  k_norm   <<<NQ / 256,   256, 0, stream>>>(patches, wsu, wsf, out);
}